// Spiking1DLayer_9895604650114
// MI455X (gfx1250) — compile-verified
//
#include <hip/hip_runtime.h>
#include <cstdint>
#include <cstddef>

// ---------------------------------------------------------------------------
// LIF spiking scan, MI455X (gfx1250).
// Bandwidth-bound streaming recurrence: 262 MB HBM traffic, ~11 us floor at
// 23.3 TB/s. No matmul -> no WMMA; CDNA5 win is the async global->LDS DMA
// path (ASYNCcnt) for deep prefetch + LDS staging for coalescing.
//
// v4: compile-time specialized full/tail chunk paths -> no runtime predicate
// or EXEC save/restore in the steady state; division-free mapping (row=it,
// quad=tid); LDS row stride 33 quads for conflict-free 2-phase b128 access.
// ---------------------------------------------------------------------------

// Problem constants (reference: B=64, C=256, T=2000)
#define T_TOTAL  2000
#define C_DIM    256
#define TB       128               // timesteps per main chunk
#define TB4      32                // float4 quads per row per main chunk
#define TB4_TAIL 20                // tail: 2000 - 15*128 = 80 steps = 20 quads
#define NCHUNK   16                // 15 main + 1 tail
#define ROWS     32                // rows per block == one wave32
#define LSTRIDE  33                // LDS row stride in quads (bank-safe pad)

// ---- CDNA5 async global->LDS copy + ASYNCcnt wait ----
#if defined(__has_builtin)
#if __has_builtin(__builtin_amdgcn_global_load_async_to_lds_b128)
#define HAVE_ASYNC_BUILTIN 1
#endif
#if __has_builtin(__builtin_amdgcn_s_wait_asynccnt)
#define HAVE_WAIT_BUILTIN 1
#endif
#endif

// Exact pointee type from the builtin's signature:
//   'int __attribute__((__vector_size__(4 * sizeof(int)))) __device__ *'
typedef int v4i __attribute__((vector_size(4 * sizeof(int))));
typedef __attribute__((address_space(1))) v4i as1_v4i;   // global (cuda_device)
typedef __attribute__((address_space(3))) v4i as3_v4i;   // LDS (cuda_shared)

__device__ __forceinline__ void async_copy_b128(const void* gsrc, void* lds_dst) {
#ifdef HAVE_ASYNC_BUILTIN
  __builtin_amdgcn_global_load_async_to_lds_b128(
      (as1_v4i*)gsrc, (as3_v4i*)lds_dst, /*offset=*/0, /*cpol=*/0);
#else
  unsigned loff = (unsigned)(uintptr_t)lds_dst;  // low 32 bits == LDS offset
  asm volatile("global_load_async_to_lds_b128 %0, %1, off"
               :: "v"(loff), "v"(gsrc)
               : "memory");
#endif
}

#ifdef HAVE_WAIT_BUILTIN
#define WAIT_ASYNC(N)                                 \
  do {                                                \
    __builtin_amdgcn_s_wait_asynccnt(N);              \
    asm volatile("" ::: "memory");                    \
  } while (0)
#else
#define WAIT_ASYNC(N) asm volatile("s_wait_asynccnt " #N ::: "memory")
#endif

// ---------------------------------------------------------------------------

// Issue one chunk's async copies: iteration 'it' moves row it's contiguous
// 512 B segment (lane tid = quad tid). PRED=true masks lanes >= QUADS (tail).
// Every chunk issues exactly ROWS=32 async instructions regardless of PRED,
// so ASYNCcnt bookkeeping is uniform (<= 64 outstanding, wait<=32 releases
// the oldest chunk; async loads retire in order).
template <bool PRED, int QUADS>
__device__ __forceinline__ void issue_chunk(const float* gbase, float4* lbase, int tid) {
  if (!PRED || tid < QUADS) {
#pragma unroll
    for (int it = 0; it < ROWS; ++it) {
      async_copy_b128(gbase + (size_t)it * T_TOTAL, lbase + it * LSTRIDE);
    }
  }
}

// Writeback one chunk's spikes: one global_store_b128 per row, constant
// immediate offsets (row=it, quad=tid).
template <bool PRED, int QUADS>
__device__ __forceinline__ void writeback_chunk(float4* obase, const float4* lbase, int tid) {
  if (!PRED || tid < QUADS) {
#pragma unroll
    for (int it = 0; it < ROWS; ++it) {
      obase[(size_t)it * (T_TOTAL / 4)] = lbase[it * LSTRIDE];
    }
  }
}

// Run NQ quads (4*NQ timesteps) of the LIF recurrence in-place on one LDS row.
template <int NQ>
__device__ __forceinline__ void lif_run_chunk(float4* myrow, float& mem, float& spk,
                                              float alpha, float beta, float vth) {
#pragma unroll
  for (int q = 0; q < NQ; ++q) {
    const float4 v = myrow[q];
    float4 s;
    mem = (mem - spk * vth) * beta + v.x * alpha;
    spk = ((mem - vth) > 0.0f) ? 1.0f : 0.0f;  s.x = spk;
    mem = (mem - spk * vth) * beta + v.y * alpha;
    spk = ((mem - vth) > 0.0f) ? 1.0f : 0.0f;  s.y = spk;
    mem = (mem - spk * vth) * beta + v.z * alpha;
    spk = ((mem - vth) > 0.0f) ? 1.0f : 0.0f;  s.z = spk;
    mem = (mem - spk * vth) * beta + v.w * alpha;
    spk = ((mem - vth) > 0.0f) ? 1.0f : 0.0f;  s.w = spk;
    myrow[q] = s;
  }
}

__global__ __launch_bounds__(ROWS) void lif_scan_kernel(
    const float* __restrict__ x,        // (B*C, T) row-major
    const float* __restrict__ alpha_p,  // scalar
    const float* __restrict__ beta_p,   // scalar
    const float* __restrict__ vth_p,    // (C,)
    float* __restrict__ out)            // (B*C, T)
{
  // Double-buffered input tile; spikes written back in place.
  // Row stride 33 quads: lanes start at bank 4*tid mod 64 -> optimal 2-phase
  // conflict-free b128 LDS access for load, compute and writeback phases.
  __shared__ float4 smem[2][ROWS * LSTRIDE];   // 2 * 32 * 33 * 16 B = 33.8 KB

  const int tid  = threadIdx.x;            // 0..31, one row per lane
  const int row0 = blockIdx.x * ROWS;      // first global row of this block
  const float alpha = alpha_p[0];
  const float beta  = beta_p[0];
  const float vth   = vth_p[(row0 + tid) & (C_DIM - 1)];  // c = row % C

  const float* gb = x + (size_t)row0 * T_TOTAL + (size_t)tid * 4;  // quad base
  float4* ob = (float4*)out + (size_t)row0 * (T_TOTAL / 4) + tid;
  float4* l0 = &smem[0][tid];
  float4* l1 = &smem[1][tid];

  // Prologue: prefetch chunks 0 and 1 (64 async b128 in flight).
  issue_chunk<false, TB4>(gb, l0, tid);
  issue_chunk<false, TB4>(gb + TB, l1, tid);

  float mem = 0.0f, spk = 0.0f;

  // 15 main chunks of 128 timesteps.
  for (int c = 0; c < NCHUNK - 1; ++c) {
    const int buf = c & 1;
    float4* lbase = buf ? l1 : l0;

    // Async loads retire in order: <=32 outstanding guarantees chunk c landed.
    WAIT_ASYNC(32);

    lif_run_chunk<TB4>(&smem[buf][tid * LSTRIDE], mem, spk, alpha, beta, vth);
    writeback_chunk<false, TB4>(ob + (size_t)c * TB4, lbase, tid);

    // Re-arm this buffer with chunk c+2 (uniform scalar branch selects the
    // tail-specialized issue). Safe: this wave's writeback ds_loads already
    // delivered into VGPRs, and the block is a single wave.
    const int next = c + 2;
    if (next < NCHUNK - 1) {
      issue_chunk<false, TB4>(gb + (size_t)next * TB, lbase, tid);
    } else if (next == NCHUNK - 1) {
      issue_chunk<true, TB4_TAIL>(gb + (size_t)next * TB, lbase, tid);
    }
  }

  // Tail chunk: t = 1920..1999 (80 steps, 20 quads), buffer (NCHUNK-1)&1 = 1.
  WAIT_ASYNC(0);
  lif_run_chunk<TB4_TAIL>(&smem[1][tid * LSTRIDE], mem, spk, alpha, beta, vth);
  writeback_chunk<true, TB4_TAIL>(ob + (size_t)(NCHUNK - 1) * TB4, l1, tid);
}

// ---------------------------------------------------------------------------

extern "C" void kernel_launch(void* const* d_in, const int* in_sizes, int n_in,
                              void* d_out, int out_size, void* d_ws, size_t ws_size,
                              hipStream_t stream) {
  (void)n_in; (void)out_size; (void)d_ws; (void)ws_size;
  const float* x     = (const float*)d_in[0];  // (B, C, T) float32
  const float* alpha = (const float*)d_in[1];  // scalar
  const float* beta  = (const float*)d_in[2];  // scalar
  const float* vth   = (const float*)d_in[3];  // (C,)
  float* out = (float*)d_out;

  const int rows = in_sizes[0] / T_TOTAL;      // B*C = 16384
  dim3 grid(rows / ROWS);                      // 512 single-wave blocks
  dim3 block(ROWS);                            // 32 threads = 1 wave32

  hipLaunchKernelGGL(lif_scan_kernel, grid, block, 0, stream,
                     x, alpha, beta, vth, out);
}